// CLIPTextEmbeddings_30116310680170
// MI455X (gfx1250) — compile-verified
//
#include <hip/hip_runtime.h>
#include <stdint.h>

// CLIP text embeddings: out[s, :] = token_weight[input_ids[s], :] + position_weight[position_ids[s], :]
// SEQ=77, DIM=768, f32 everywhere. Pure gather+add: memory path only, no matrix FLOPs.

#define DIM 768
#define VEC4 (DIM / 4)   // 192 float4 chunks per row -> 192 threads per block

__global__ __launch_bounds__(VEC4)
void clip_embed_kernel(const int* __restrict__ input_ids,
                       const int* __restrict__ position_ids,
                       const float* __restrict__ token_w,
                       const float* __restrict__ pos_w,
                       float* __restrict__ out)
{
    __shared__ float4 smem_tok[VEC4];
    __shared__ float4 smem_pos[VEC4];

    const int s = blockIdx.x;    // sequence position 0..76
    const int t = threadIdx.x;   // 0..191 (6 wave32s)

    // Uniform per block -> compiler lowers these to scalar loads.
    const int tok = input_ids[s];
    const int pos = position_ids[s];

    const float4* tok_row = (const float4*)(token_w + (size_t)tok * DIM);
    const float4* pos_row = (const float4*)(pos_w   + (size_t)pos * DIM);

    const float4* ga = tok_row + t;   // 16B per lane, coalesced B128
    const float4* gb = pos_row + t;

    // LDS byte offsets: low 32 bits of the generic (flat) pointer to a
    // __shared__ object are the LDS address (aperture lives in the high bits).
    unsigned lds_tok = (unsigned)(uintptr_t)(&smem_tok[t]);
    unsigned lds_pos = (unsigned)(uintptr_t)(&smem_pos[t]);

    // CDNA5 async global->LDS copies (tracked by ASYNCcnt). Each lane stages
    // its own 16B chunk of both gathered rows directly into LDS, bypassing
    // the VGPR load-return path.
    asm volatile("global_load_async_to_lds_b128 %0, %1, off"
                 :: "v"(lds_tok), "v"(ga) : "memory");
    asm volatile("global_load_async_to_lds_b128 %0, %1, off"
                 :: "v"(lds_pos), "v"(gb) : "memory");
    // Each wave waits only on its own async ops; it reads back exactly the
    // LDS region its own lanes staged, so no workgroup barrier is required.
    asm volatile("s_wait_asynccnt 0" ::: "memory");

    float4 a = smem_tok[t];
    float4 b = smem_pos[t];
    float4 r;
    r.x = a.x + b.x;
    r.y = a.y + b.y;
    r.z = a.z + b.z;
    r.w = a.w + b.w;

    // Coalesced B128 store of the output row.
    ((float4*)(out + (size_t)s * DIM))[t] = r;
}

extern "C" void kernel_launch(void* const* d_in, const int* in_sizes, int n_in,
                              void* d_out, int out_size, void* d_ws, size_t ws_size,
                              hipStream_t stream) {
    const int*   input_ids    = (const int*)d_in[0];
    const int*   position_ids = (const int*)d_in[1];
    const float* token_w      = (const float*)d_in[2];
    const float* pos_w        = (const float*)d_in[3];
    float*       out          = (float*)d_out;

    const int seq = in_sizes[0];  // 77
    clip_embed_kernel<<<dim3(seq), dim3(VEC4), 0, stream>>>(
        input_ids, position_ids, token_w, pos_w, out);
}